// ChannelCompressor_11862699671669
// MI455X (gfx1250) — compile-verified
//
#include <hip/hip_runtime.h>
#include <math.h>

typedef __attribute__((ext_vector_type(2))) float v2f;
typedef __attribute__((ext_vector_type(8))) float v8f;

#define BATCH   16
#define SEQ     4096
#define D_MODEL 1024
#define HIDDEN  128
#define NS      32                    // seq chunks for the pooling tree-reduce
#define ROWS_PER_CHUNK (SEQ / NS)     // 128

// ---------------------------------------------------------------------------
// Kernel 1: partial sums over the sequence dimension.
// Block = (b, seq-chunk). 256 threads cover one 1024-float row as float4s,
// so each loop iteration is one fully-coalesced 4KB row read.
// ---------------------------------------------------------------------------
__global__ __launch_bounds__(256) void pool_partial(const float* __restrict__ x,
                                                    float* __restrict__ partial) {
    const int b  = blockIdx.x / NS;
    const int sc = blockIdx.x % NS;
    const int t  = threadIdx.x;                       // 0..255 -> d4 index
    const float4* xr = (const float4*)x +
        (size_t)(b * SEQ + sc * ROWS_PER_CHUNK) * (D_MODEL / 4);
    float4 acc = make_float4(0.f, 0.f, 0.f, 0.f);
    for (int r = 0; r < ROWS_PER_CHUNK; ++r) {
        float4 v = xr[(size_t)r * (D_MODEL / 4) + t];
        acc.x += v.x; acc.y += v.y; acc.z += v.z; acc.w += v.w;
    }
    ((float4*)partial)[(size_t)blockIdx.x * (D_MODEL / 4) + t] = acc;
}

// ---------------------------------------------------------------------------
// Kernel 2: fold the NS partials into pooled[b][d] = mean over S.
// ---------------------------------------------------------------------------
__global__ __launch_bounds__(256) void pool_reduce(const float* __restrict__ partial,
                                                   float* __restrict__ pooled) {
    const int i = blockIdx.x * 256 + threadIdx.x;     // 0 .. 16*1024-1
    const int b = i / D_MODEL;
    const int d = i % D_MODEL;
    float s = 0.f;
    for (int sc = 0; sc < NS; ++sc)
        s += partial[(size_t)(b * NS + sc) * D_MODEL + d];
    pooled[i] = s * (1.0f / (float)SEQ);
}

// ---------------------------------------------------------------------------
// Kernel 3: h = gelu(pooled @ W1 + b1) via V_WMMA_F32_16X16X4_F32.
// One block, 8 waves; wave w owns columns [16w, 16w+16) of HIDDEN.
// M=16 == batch, K=1024, 256 WMMA steps per wave, fp32 accumulate.
// ---------------------------------------------------------------------------
__global__ __launch_bounds__(256) void gemm1_gelu(const float* __restrict__ pooled,
                                                  const float* __restrict__ W1,
                                                  const float* __restrict__ b1,
                                                  float* __restrict__ h) {
    const int tid   = threadIdx.x;
    const int wave  = tid >> 5;
    const int lane  = tid & 31;
    const int m     = lane & 15;            // A: M row / B,D: N within tile
    const int khalf = (lane >> 4) << 1;     // 0 for lanes 0-15, 2 for 16-31
    const int n     = wave * 16 + m;        // global column in HIDDEN

    v8f c = {};
    for (int k = 0; k < D_MODEL; k += 4) {
        const int kk = k + khalf;
        v2f a, bm;
        a.x  = pooled[m * D_MODEL + kk];
        a.y  = pooled[m * D_MODEL + kk + 1];
        bm.x = W1[(size_t)kk * HIDDEN + n];
        bm.y = W1[(size_t)(kk + 1) * HIDDEN + n];
        c = __builtin_amdgcn_wmma_f32_16x16x4_f32(false, a, false, bm,
                                                  (short)0, c, false, false);
    }
    const int hi = lane >> 4;
    const float bias = b1[n];
#pragma unroll
    for (int v = 0; v < 8; ++v) {
        const int M = v + 8 * hi;
        const float pre = c[v] + bias;
        // exact (erf) GELU, matching approximate=False
        h[M * HIDDEN + n] = 0.5f * pre * (1.0f + erff(pre * 0.70710678118654752f));
    }
}

// ---------------------------------------------------------------------------
// Kernel 4: scale = sigmoid(h @ W2 + b2). M=16, K=128, N=1024.
// 8 blocks x 8 waves = 64 N-tiles of 16; 32 WMMA steps per wave.
// ---------------------------------------------------------------------------
__global__ __launch_bounds__(256) void gemm2_sigmoid(const float* __restrict__ h,
                                                     const float* __restrict__ W2,
                                                     const float* __restrict__ b2,
                                                     float* __restrict__ scale) {
    const int tid   = threadIdx.x;
    const int wave  = tid >> 5;
    const int lane  = tid & 31;
    const int m     = lane & 15;
    const int khalf = (lane >> 4) << 1;
    const int n     = (blockIdx.x * 8 + wave) * 16 + m;   // column in D_MODEL

    v8f c = {};
    for (int k = 0; k < HIDDEN; k += 4) {
        const int kk = k + khalf;
        v2f a, bm;
        a.x  = h[m * HIDDEN + kk];
        a.y  = h[m * HIDDEN + kk + 1];
        bm.x = W2[(size_t)kk * D_MODEL + n];
        bm.y = W2[(size_t)(kk + 1) * D_MODEL + n];
        c = __builtin_amdgcn_wmma_f32_16x16x4_f32(false, a, false, bm,
                                                  (short)0, c, false, false);
    }
    const int hi = lane >> 4;
    const float bias = b2[n];
#pragma unroll
    for (int v = 0; v < 8; ++v) {
        const int M = v + 8 * hi;
        const float pre = c[v] + bias;
        scale[M * D_MODEL + n] = 1.0f / (1.0f + expf(-pre));
    }
}

// ---------------------------------------------------------------------------
// Kernel 5: out = x * scale[b][d], broadcast over S. Dominant traffic:
// 256 MiB read + 256 MiB write, float4 (global_load/store_b128), grid-stride.
// ---------------------------------------------------------------------------
__global__ __launch_bounds__(256) void apply_scale(const float4* __restrict__ x4,
                                                   const float4* __restrict__ scale4,
                                                   float4* __restrict__ out4) {
    const int R = D_MODEL / 4;                               // 256 float4 per row
    const size_t total = (size_t)BATCH * SEQ * R;            // 16M float4
    for (size_t i = (size_t)blockIdx.x * blockDim.x + threadIdx.x; i < total;
         i += (size_t)gridDim.x * blockDim.x) {
        const size_t b  = i / ((size_t)SEQ * R);
        const int    d4 = (int)(i & (size_t)(R - 1));
        const float4 s  = scale4[b * R + d4];
        float4 v = x4[i];
        v.x *= s.x; v.y *= s.y; v.z *= s.z; v.w *= s.w;
        out4[i] = v;
    }
}

// ---------------------------------------------------------------------------
extern "C" void kernel_launch(void* const* d_in, const int* in_sizes, int n_in,
                              void* d_out, int out_size, void* d_ws, size_t ws_size,
                              hipStream_t stream) {
    const float* x  = (const float*)d_in[0];
    const float* W1 = (const float*)d_in[1];
    const float* b1 = (const float*)d_in[2];
    const float* W2 = (const float*)d_in[3];
    const float* b2 = (const float*)d_in[4];
    float* out = (float*)d_out;

    // workspace layout (all float, ~2.14 MB total)
    float* partial = (float*)d_ws;                                 // 16*32*1024
    float* pooled  = partial + (size_t)BATCH * NS * D_MODEL;       // 16*1024
    float* h       = pooled + BATCH * D_MODEL;                     // 16*128
    float* scale   = h + BATCH * HIDDEN;                           // 16*1024

    pool_partial <<<BATCH * NS,            256, 0, stream>>>(x, partial);
    pool_reduce  <<<BATCH * D_MODEL / 256, 256, 0, stream>>>(partial, pooled);
    gemm1_gelu   <<<1,                     256, 0, stream>>>(pooled, W1, b1, h);
    gemm2_sigmoid<<<D_MODEL / HIDDEN,      256, 0, stream>>>(h, W2, b2, scale);
    apply_scale  <<<8192,                  256, 0, stream>>>((const float4*)x,
                                                             (const float4*)scale,
                                                             (float4*)out);
}